// GCN_85358180041225
// MI455X (gfx1250) — compile-verified
//
#include <hip/hip_runtime.h>
#include <hip/hip_bf16.h>

// ---------------------------------------------------------------------------
// GCN forward for MI455X (gfx1250).  fp32 end-to-end (matches reference
// numerics); dense transforms via V_WMMA_F32_16X16X4_F32; edge scatter via
// hardware fp32 global atomics (destination fits in 192MB L2).
// ---------------------------------------------------------------------------

typedef float v2f __attribute__((ext_vector_type(2)));
typedef float v8f __attribute__((ext_vector_type(8)));

#define F_IN 128
#define DIM  64
#define NCLS 10
#define NGR  512   // num_graphs (scalar in reference = 512)
#define EPS  1e-5f

// ---------------------------------------------------------------------------
// 0) zero scratch regions used with atomics
// ---------------------------------------------------------------------------
__global__ void zero_kernel(unsigned* __restrict__ deg, float* __restrict__ stats,
                            float* __restrict__ pooled, float* __restrict__ cnt, int N) {
  int i = blockIdx.x * 256 + threadIdx.x;
  if (i < N) deg[i] = 0u;
  if (i < 2 * DIM) stats[i] = 0.0f;
  if (i < NGR * DIM) pooled[i] = 0.0f;
  if (i < NGR) cnt[i] = 0.0f;
}

// ---------------------------------------------------------------------------
// 1) in-degree (targets) via integer atomics
// ---------------------------------------------------------------------------
__global__ void deg_kernel(const long long* __restrict__ ei, unsigned* __restrict__ deg,
                           long long E) {
  long long e = (long long)blockIdx.x * 256 + threadIdx.x;
  if (e < E) atomicAdd(&deg[ei[E + e]], 1u);   // col = ei[1][e]
}

// 2) dis = rsqrt(deg+1), dis2 = 1/(deg+1)   (+1 = self loop)
__global__ void dis_kernel(const unsigned* __restrict__ deg, float* __restrict__ dis,
                           float* __restrict__ dis2, int N) {
  int i = blockIdx.x * 256 + threadIdx.x;
  if (i < N) {
    float d = (float)(deg[i] + 1u);
    dis[i]  = rsqrtf(d);
    dis2[i] = 1.0f / d;
  }
}

// ---------------------------------------------------------------------------
// 3) WMMA GEMM:  xl = A' @ W   where A' = A          (layer 1)
//                              or    A' = A*scale+shift (BN affine fused, L2-4)
//    epilogue also writes agg = dis2[row] * xl  (self-loop init of scatter acc)
//    Block = 256 threads = 8 waves; each wave owns one 16x16 output tile:
//    block covers 32 rows x 64 cols.  W staged in LDS.
// ---------------------------------------------------------------------------
template <int K, bool AFFINE>
__global__ void gcn_gemm(const float* __restrict__ A, const float* __restrict__ W,
                         const float* __restrict__ scale, const float* __restrict__ shift,
                         const float* __restrict__ dis2,
                         float* __restrict__ xl, float* __restrict__ agg, int N) {
  __shared__ float sW[K * DIM];
  for (int i = threadIdx.x; i < K * DIM; i += blockDim.x) sW[i] = W[i];
  __syncthreads();

  const int wave = threadIdx.x >> 5;   // 0..7
  const int lane = threadIdx.x & 31;
  const int rowTile = blockIdx.x * 2 + (wave >> 2);
  const int colTile = wave & 3;
  const int m0 = rowTile * 16;
  const int n0 = colTile * 16;
  if (m0 >= N) return;

  // A fragment (16x4 fp32): lanes 0-15 -> K={0,1}, lanes 16-31 -> K={2,3}
  const int mA   = lane & 15;
  const int kgrp = (lane >> 4) * 2;
  const int nB   = lane & 15;
  int mrow = m0 + mA; if (mrow >= N) mrow = N - 1;
  const float* Arow = A + (long long)mrow * K;

  v8f c = {0.f, 0.f, 0.f, 0.f, 0.f, 0.f, 0.f, 0.f};
  #pragma unroll
  for (int k = 0; k < K; k += 4) {
    float a0 = Arow[k + kgrp];
    float a1 = Arow[k + kgrp + 1];
    if (AFFINE) {
      a0 = a0 * scale[k + kgrp]     + shift[k + kgrp];
      a1 = a1 * scale[k + kgrp + 1] + shift[k + kgrp + 1];
    }
    v2f a; a.x = a0; a.y = a1;
    v2f b;
    b.x = sW[(k + kgrp)     * DIM + n0 + nB];
    b.y = sW[(k + kgrp + 1) * DIM + n0 + nB];
    c = __builtin_amdgcn_wmma_f32_16x16x4_f32(false, a, false, b, (short)0, c,
                                              false, false);
  }

  // C/D layout: VGPR v, lane L: row = v + 8*(L>>4), col = L&15
  const int half = lane >> 4;
  const int n    = lane & 15;
  #pragma unroll
  for (int v = 0; v < 8; ++v) {
    int m = m0 + v + 8 * half;
    if (m < N) {
      long long o = (long long)m * DIM + n0 + n;
      float val = c[v];
      xl[o]  = val;
      agg[o] = dis2[m] * val;        // self-loop: norm = dis^2 = 1/deg
    }
  }
}

// ---------------------------------------------------------------------------
// 4) edge scatter: agg[col] += dis[row]*dis[col] * xl[row]
//    16 threads per edge, each moves a float4 (global_load_b128 + 4 fp32 atomics)
// ---------------------------------------------------------------------------
__global__ void edge_kernel(const long long* __restrict__ ei, const float* __restrict__ xl,
                            const float* __restrict__ dis, float* __restrict__ agg,
                            long long E) {
  const int slot  = threadIdx.x >> 4;           // 16 edges / block
  const int chunk = threadIdx.x & 15;           // 16 x float4 = 64 features
  long long e = (long long)blockIdx.x * 16 + slot;
  if (e >= E) return;
  long long r = ei[e];
  long long cl = ei[E + e];
  float nrm = dis[r] * dis[cl];
  const float4 v = *(const float4*)(xl + r * DIM + chunk * 4);
  float* dst = agg + cl * DIM + chunk * 4;
  unsafeAtomicAdd(dst + 0, v.x * nrm);
  unsafeAtomicAdd(dst + 1, v.y * nrm);
  unsafeAtomicAdd(dst + 2, v.z * nrm);
  unsafeAtomicAdd(dst + 3, v.w * nrm);
}

// ---------------------------------------------------------------------------
// 5) h = relu(agg + b); accumulate per-feature sum/sumsq (BN batch stats)
//    block = 256 threads = 64 features x 4 row-groups; 128 rows per block
// ---------------------------------------------------------------------------
__global__ void relu_stats_kernel(const float* __restrict__ agg, const float* __restrict__ b,
                                  float* __restrict__ h, float* __restrict__ stats, int N) {
  __shared__ float ssum[256];
  __shared__ float ssq[256];
  const int f  = threadIdx.x & 63;
  const int rg = threadIdx.x >> 6;              // 0..3
  const int base = blockIdx.x * 128;
  const float bf = b[f];
  float s = 0.f, q = 0.f;
  for (int r = rg; r < 128; r += 4) {
    int row = base + r;
    if (row < N) {
      long long o = (long long)row * DIM + f;
      float v = fmaxf(agg[o] + bf, 0.0f);
      h[o] = v;
      s += v; q += v * v;
    }
  }
  ssum[threadIdx.x] = s; ssq[threadIdx.x] = q;
  __syncthreads();
  if (rg == 0) {
    s = ssum[f] + ssum[64 + f] + ssum[128 + f] + ssum[192 + f];
    q = ssq[f]  + ssq[64 + f]  + ssq[128 + f]  + ssq[192 + f];
    unsafeAtomicAdd(&stats[f], s);
    unsafeAtomicAdd(&stats[DIM + f], q);
  }
}

// 6) BN finalize: scale = g*rsqrt(var+eps), shift = be - mean*scale; rezero stats
__global__ void bn_finalize_kernel(const float* __restrict__ g, const float* __restrict__ be,
                                   float* __restrict__ stats, float* __restrict__ scale,
                                   float* __restrict__ shift, float invN) {
  int f = threadIdx.x;   // 64 threads
  float mean = stats[f] * invN;
  float var  = stats[DIM + f] * invN - mean * mean;
  float inv  = rsqrtf(var + EPS);
  float sc   = g[f] * inv;
  scale[f] = sc;
  shift[f] = be[f] - mean * sc;
  stats[f] = 0.f; stats[DIM + f] = 0.f;       // ready for next layer
}

// ---------------------------------------------------------------------------
// 7) segment-sum pooling of BN(h4) keyed on batch[]
// ---------------------------------------------------------------------------
__global__ void pool_kernel(const float* __restrict__ h, const float* __restrict__ scale,
                            const float* __restrict__ shift, const long long* __restrict__ batch,
                            float* __restrict__ pooled, float* __restrict__ cnt, int N) {
  const int slot  = threadIdx.x >> 4;
  const int chunk = threadIdx.x & 15;
  int i = blockIdx.x * 16 + slot;
  if (i >= N) return;
  long long g = batch[i];
  int f = chunk * 4;
  const float4 v  = *(const float4*)(h + (long long)i * DIM + f);
  const float4 sc = *(const float4*)(scale + f);
  const float4 sh = *(const float4*)(shift + f);
  float* dst = pooled + g * DIM + f;
  unsafeAtomicAdd(dst + 0, v.x * sc.x + sh.x);
  unsafeAtomicAdd(dst + 1, v.y * sc.y + sh.y);
  unsafeAtomicAdd(dst + 2, v.z * sc.z + sh.z);
  unsafeAtomicAdd(dst + 3, v.w * sc.w + sh.w);
  if (chunk == 0) unsafeAtomicAdd(&cnt[g], 1.0f);
}

// 8) head: mean pool -> FC(64->10) -> log_softmax
__global__ void head_kernel(const float* __restrict__ pooled, const float* __restrict__ cnt,
                            const float* __restrict__ fcW, const float* __restrict__ fcb,
                            float* __restrict__ out, int G) {
  __shared__ float sW[DIM * NCLS];
  __shared__ float sb[NCLS];
  for (int i = threadIdx.x; i < DIM * NCLS; i += blockDim.x) sW[i] = fcW[i];
  if (threadIdx.x < NCLS) sb[threadIdx.x] = fcb[threadIdx.x];
  __syncthreads();
  for (int g = threadIdx.x; g < G; g += blockDim.x) {
    float c = fmaxf(cnt[g], 1.0f);
    float inv = 1.0f / c;
    float logits[NCLS];
    #pragma unroll
    for (int k = 0; k < NCLS; ++k) logits[k] = sb[k];
    for (int d = 0; d < DIM; ++d) {
      float p = pooled[g * DIM + d] * inv;
      #pragma unroll
      for (int k = 0; k < NCLS; ++k) logits[k] += p * sW[d * NCLS + k];
    }
    float m = logits[0];
    #pragma unroll
    for (int k = 1; k < NCLS; ++k) m = fmaxf(m, logits[k]);
    float sum = 0.f;
    #pragma unroll
    for (int k = 0; k < NCLS; ++k) sum += __expf(logits[k] - m);
    float lse = m + __logf(sum);
    #pragma unroll
    for (int k = 0; k < NCLS; ++k) out[g * NCLS + k] = logits[k] - lse;
  }
}

// ---------------------------------------------------------------------------
static inline char* wsAlign(char*& p, size_t bytes) {
  char* r = p;
  p += (bytes + 255) & ~(size_t)255;
  return r;
}

extern "C" void kernel_launch(void* const* d_in, const int* in_sizes, int n_in,
                              void* d_out, int out_size, void* d_ws, size_t ws_size,
                              hipStream_t stream) {
  const float*     x     = (const float*)d_in[0];
  const long long* ei    = (const long long*)d_in[1];
  const long long* batch = (const long long*)d_in[2];
  const float* W[4]  = {(const float*)d_in[4], (const float*)d_in[6],
                        (const float*)d_in[8], (const float*)d_in[10]};
  const float* B[4]  = {(const float*)d_in[5], (const float*)d_in[7],
                        (const float*)d_in[9], (const float*)d_in[11]};
  const float* Ga[4] = {(const float*)d_in[12], (const float*)d_in[14],
                        (const float*)d_in[16], (const float*)d_in[18]};
  const float* Be[4] = {(const float*)d_in[13], (const float*)d_in[15],
                        (const float*)d_in[17], (const float*)d_in[19]};
  const float* fcW = (const float*)d_in[20];
  const float* fcb = (const float*)d_in[21];

  const int       N = in_sizes[0] / F_IN;    // 100000
  const long long E = in_sizes[1] / 2;       // 3200000
  const int       G = NGR;                   // 512 (reference constant)

  // ---- workspace carve-out ----
  char* p = (char*)d_ws;
  unsigned* deg   = (unsigned*)wsAlign(p, (size_t)N * 4);
  float* dis      = (float*)wsAlign(p, (size_t)N * 4);
  float* dis2     = (float*)wsAlign(p, (size_t)N * 4);
  float* xl       = (float*)wsAlign(p, (size_t)N * DIM * 4);
  float* agg      = (float*)wsAlign(p, (size_t)N * DIM * 4);
  float* h        = (float*)wsAlign(p, (size_t)N * DIM * 4);
  float* stats    = (float*)wsAlign(p, 2 * DIM * 4);
  float* scale    = (float*)wsAlign(p, DIM * 4);
  float* shift    = (float*)wsAlign(p, DIM * 4);
  float* pooled   = (float*)wsAlign(p, (size_t)G * DIM * 4);
  float* cnt      = (float*)wsAlign(p, (size_t)G * 4);

  const dim3 blk(256);
  const int  nBlkN     = (N + 255) / 256;
  const int  nBlkEdge  = (int)((E + 15) / 16);
  const int  nBlkGemm  = ((N + 15) / 16 + 1) / 2;           // 2 row-tiles/block
  const int  nBlkStats = (N + 127) / 128;
  const int  nBlkPool  = (N + 15) / 16;
  const float invN = 1.0f / (float)N;

  // prologue: zero atomics targets, degrees, normalization
  zero_kernel<<<nBlkN, blk, 0, stream>>>(deg, stats, pooled, cnt, N);
  deg_kernel<<<(int)((E + 255) / 256), blk, 0, stream>>>(ei, deg, E);
  dis_kernel<<<nBlkN, blk, 0, stream>>>(deg, dis, dis2, N);

  // ---- layer 1 (K = 128, no affine on input) ----
  gcn_gemm<F_IN, false><<<nBlkGemm, blk, 0, stream>>>(x, W[0], scale, shift, dis2, xl, agg, N);
  edge_kernel<<<nBlkEdge, blk, 0, stream>>>(ei, xl, dis, agg, E);
  relu_stats_kernel<<<nBlkStats, blk, 0, stream>>>(agg, B[0], h, stats, N);
  bn_finalize_kernel<<<1, DIM, 0, stream>>>(Ga[0], Be[0], stats, scale, shift, invN);

  // ---- layers 2..4 (K = 64, BN affine fused into A load) ----
  for (int l = 1; l < 4; ++l) {
    gcn_gemm<DIM, true><<<nBlkGemm, blk, 0, stream>>>(h, W[l], scale, shift, dis2, xl, agg, N);
    edge_kernel<<<nBlkEdge, blk, 0, stream>>>(ei, xl, dis, agg, E);
    relu_stats_kernel<<<nBlkStats, blk, 0, stream>>>(agg, B[l], h, stats, N);
    bn_finalize_kernel<<<1, DIM, 0, stream>>>(Ga[l], Be[l], stats, scale, shift, invN);
  }

  // ---- pooling + head ----
  pool_kernel<<<nBlkPool, blk, 0, stream>>>(h, scale, shift, batch, pooled, cnt, N);
  head_kernel<<<1, blk, 0, stream>>>(pooled, cnt, fcW, fcb, (float*)d_out, G);
}